// ContextGating_22101901705855
// MI455X (gfx1250) — compile-verified
//
#include <hip/hip_runtime.h>
#include <float.h>

#define HID    512
#define AGENTS 128
#define BATCH  512
#define MROWS  (BATCH * AGENTS)   // 65536 rows in the big GEMM
#define NT     8                  // N-tiles (16 cols each) per wave -> 128 cols/wave

typedef __attribute__((ext_vector_type(2))) float v2f;
typedef __attribute__((ext_vector_type(8))) float v8f;

// ---------------------------------------------------------------------------
// Kernel 1: context = context_embedding @ W_c^T + b_c      (512 x 512)
// One wave computes a 16x128 tile via V_WMMA_F32_16X16X4_F32.
// ---------------------------------------------------------------------------
__global__ __launch_bounds__(128) void ctx_gemm_kernel(
    const float* __restrict__ ce, const float* __restrict__ Wc,
    const float* __restrict__ bc, float* __restrict__ ctx) {
  const int wid  = threadIdx.x >> 5;
  const int lane = threadIdx.x & 31;
  const int job  = blockIdx.x * 4 + wid;     // 32 mtiles * 4 ngroups = 128 jobs
  const int mtile  = job >> 2;
  const int ngroup = job & 3;
  const int r  = lane & 15;                  // row-in-tile (M for A, N for B)
  const int kh = lane >> 4;                  // K half: lanes 16-31 hold K+2,K+3
  const int m0 = mtile * 16;
  const int n0 = ngroup * (16 * NT);

  const float* aptr = ce + (size_t)(m0 + r) * HID + 2 * kh;
  const float* bptr = Wc + (size_t)(n0 + r) * HID + 2 * kh;

  v8f acc[NT];
#pragma unroll
  for (int t = 0; t < NT; ++t) acc[t] = (v8f){0.f, 0.f, 0.f, 0.f, 0.f, 0.f, 0.f, 0.f};

  for (int k0 = 0; k0 < HID; k0 += 4) {
    v2f a = *(const v2f*)(aptr + k0);
#pragma unroll
    for (int t = 0; t < NT; ++t) {
      v2f b = *(const v2f*)(bptr + t * 16 * HID + k0);
      acc[t] = __builtin_amdgcn_wmma_f32_16x16x4_f32(
          /*neg_a=*/false, a, /*neg_b=*/false, b,
          /*c_mod=*/(short)0, acc[t], /*reuse_a=*/false, /*reuse_b=*/false);
    }
  }

#pragma unroll
  for (int t = 0; t < NT; ++t) {
    const int n = n0 + t * 16 + r;
    const float bias = bc[n];
#pragma unroll
    for (int i = 0; i < 8; ++i) {
      const int row = m0 + i + 8 * kh;       // C/D layout: VGPR i holds M=i / M=i+8
      ctx[(size_t)row * HID + n] = acc[t][i] + bias;
    }
  }
}

// ---------------------------------------------------------------------------
// Kernel 2: embeddings = (hidden @ W_f^T + b_f) * ctx[b]   (65536 x 512)
// 4096 blocks x 4 waves; waves in a block share the same A tile (mtile).
// Each 16-row M-tile lies inside ONE batch element (128 agents % 16 == 0).
// ---------------------------------------------------------------------------
__global__ __launch_bounds__(128) void main_gemm_kernel(
    const float* __restrict__ hidden, const float* __restrict__ Wf,
    const float* __restrict__ bf, const float* __restrict__ ctx,
    float* __restrict__ emb) {
  const int wid  = threadIdx.x >> 5;
  const int lane = threadIdx.x & 31;
  const int job  = blockIdx.x * 4 + wid;     // 4096 mtiles * 4 ngroups
  const int mtile  = job >> 2;
  const int ngroup = job & 3;
  const int r  = lane & 15;
  const int kh = lane >> 4;
  const int m0 = mtile * 16;
  const int bidx = m0 >> 7;                  // batch element for this whole tile
  const int n0 = ngroup * (16 * NT);

  const float* aptr = hidden + (size_t)(m0 + r) * HID + 2 * kh;
  const float* bptr = Wf     + (size_t)(n0 + r) * HID + 2 * kh;

  v8f acc[NT];
#pragma unroll
  for (int t = 0; t < NT; ++t) acc[t] = (v8f){0.f, 0.f, 0.f, 0.f, 0.f, 0.f, 0.f, 0.f};

  for (int k0 = 0; k0 < HID; k0 += 4) {
    v2f a = *(const v2f*)(aptr + k0);
#pragma unroll
    for (int t = 0; t < NT; ++t) {
      v2f b = *(const v2f*)(bptr + t * 16 * HID + k0);
      acc[t] = __builtin_amdgcn_wmma_f32_16x16x4_f32(
          false, a, false, b, (short)0, acc[t], false, false);
    }
  }

#pragma unroll
  for (int t = 0; t < NT; ++t) {
    const int n = n0 + t * 16 + r;
    const float bias = bf[n];
    const float gate = ctx[(size_t)bidx * HID + n];
#pragma unroll
    for (int i = 0; i < 8; ++i) {
      const int row = m0 + i + 8 * kh;
      emb[(size_t)row * HID + n] = (acc[t][i] + bias) * gate;
    }
  }
}

// ---------------------------------------------------------------------------
// Kernel 3: context_out[b][n] = max over available agents of emb[b][a][n]
// Embeddings are L2-resident (134 MB < 192 MB L2); coalesced over n.
// ---------------------------------------------------------------------------
__global__ __launch_bounds__(256) void agent_max_kernel(
    const float* __restrict__ emb, const int* __restrict__ avail,
    float* __restrict__ out) {
  const int idx = blockIdx.x * 256 + threadIdx.x;    // 512*512 outputs
  const int b = idx >> 9;
  const int n = idx & (HID - 1);
  const float* base = emb + (size_t)b * AGENTS * HID + n;
  const int* av = avail + b * AGENTS;
  float m = -FLT_MAX;                                // jnp.finfo(f32).min
  for (int a = 0; a < AGENTS; ++a) {
    float v = base[(size_t)a * HID];
    if (av[a] != 0) m = fmaxf(m, v);
  }
  out[idx] = m;
}

// ---------------------------------------------------------------------------
extern "C" void kernel_launch(void* const* d_in, const int* in_sizes, int n_in,
                              void* d_out, int out_size, void* d_ws, size_t ws_size,
                              hipStream_t stream) {
  const float* hidden = (const float*)d_in[0];   // (512,128,512)
  const float* ce     = (const float*)d_in[1];   // (512,512)
  const int*   avail  = (const int*)  d_in[2];   // (512,128) bool->int
  const float* Wf     = (const float*)d_in[3];   // (512,512)
  const float* bf     = (const float*)d_in[4];   // (512,)
  const float* Wc     = (const float*)d_in[5];   // (512,512)
  const float* bc     = (const float*)d_in[6];   // (512,)

  float* emb     = (float*)d_out;                         // 65536*512 floats
  float* ctx_out = emb + (size_t)MROWS * HID;             // 512*512 floats
  float* ctx_ws  = (float*)d_ws;                          // 1 MB scratch

  // 1) context GEMM:  32 mtiles * 4 ngroups = 128 wave-jobs / 4 = 32 blocks
  ctx_gemm_kernel<<<32, 128, 0, stream>>>(ce, Wc, bc, ctx_ws);

  // 2) main GEMM + fused bias/gate epilogue: 4096 blocks of 4 waves
  main_gemm_kernel<<<4096, 128, 0, stream>>>(hidden, Wf, bf, ctx_ws, emb);

  // 3) masked per-batch max over agents: 512*512/256 = 1024 blocks
  agent_max_kernel<<<(BATCH * HID) / 256, 256, 0, stream>>>(emb, avail, ctx_out);
}